// GConvGRUCore_23330262352454
// MI455X (gfx1250) — compile-verified
//
#include <hip/hip_runtime.h>
#include <hip/hip_bf16.h>
#include <math.h>

// ---------------- problem constants ----------------
constexpr int NROW = 8192;   // graph nodes
constexpr int KDIM = 8192;   // inner dim of L @ M
constexpr int CC   = 32;     // channels (C_IN == C_OUT == 32)
constexpr int SZ   = NROW * CC;      // elements of one (N,C) matrix

// ---------------- tiling ----------------
constexpr int BR     = 128;  // rows of output per block (8 waves x 16 rows)
constexpr int KC     = 64;   // K-chunk staged in LDS
constexpr int KSPLIT = 4;    // K split across blockIdx.y (partials + reduce)
constexpr int LDA    = 68;   // padded LDS stride for L tile (conflict-free b64 reads)
constexpr int LDK    = 68;   // padded k-stride for transposed M tile (bank = 4n)

typedef float v2f __attribute__((ext_vector_type(2)));
typedef float v8f __attribute__((ext_vector_type(8)));

// D = A(16x4 f32) * B(4x16 f32) + C(16x16 f32), wave32, exact f32
__device__ __forceinline__ v8f wmma_f32(v2f a, v2f b, v8f c) {
  return __builtin_amdgcn_wmma_f32_16x16x4_f32(
      /*neg_a=*/false, a, /*neg_b=*/false, b,
      /*c_mod=*/(short)0, c, /*reuse_a=*/false, /*reuse_b=*/false);
}

// CDNA5 async copy: global -> LDS, 16B per lane, tracked by ASYNCcnt.
__device__ __forceinline__ void async_copy_b128(uint32_t lds_off, const float* g) {
  asm volatile("global_load_async_to_lds_b128 %0, %1, off"
               :: "v"(lds_off), "v"(g) : "memory");
}
__device__ __forceinline__ void wait_async0() {
  asm volatile("s_wait_asynccnt 0" ::: "memory");
}

// ------------------------------------------------------------------
// bigmm<NMAT>: partial[y] = L[:, yK:(y+1)K] @ M  for NMAT right-hand sides.
// Block: 256 threads = 8 waves; each wave owns 16 rows x 32 cols per RHS.
// L tiles: double-buffered LDS via GLOBAL_LOAD_ASYNC_TO_LDS_B128.
// M tiles: loaded + transposed into LDS so B frags are single ds_load_b64.
// ------------------------------------------------------------------
template <int NMAT>
__global__ void __launch_bounds__(256)
bigmm(const float* __restrict__ L,
      const float* __restrict__ M0, const float* __restrict__ M1,
      float* __restrict__ P0, float* __restrict__ P1) {
  __shared__ __align__(16) float sA[2][BR * LDA];
  __shared__ __align__(16) float sBT[2][NMAT][CC * LDK];   // [n][k] transposed

  const int tid    = threadIdx.x;
  const int lane   = tid & 31;
  const int wave   = tid >> 5;
  const int rowBlk = blockIdx.x * BR;
  const int kBase  = blockIdx.y * (KDIM / KSPLIT);
  const int KEND   = KDIM / KSPLIT;

  const int mrow  = lane & 15;            // A-frag row (M index)
  const int khalf = (lane >> 4) << 1;     // lanes 16-31 hold k+2,k+3
  const int bn    = lane & 15;            // B-frag column

  const uint32_t sAoff = (uint32_t)(uintptr_t)&sA[0][0];

  v8f acc[NMAT][2];
#pragma unroll
  for (int m = 0; m < NMAT; ++m) { acc[m][0] = (v8f){}; acc[m][1] = (v8f){}; }

  auto stage = [&](int buf, int kOff) {
    // L tile (BR x KC): async straight to LDS, 8 x b128 per thread
#pragma unroll
    for (int it = 0; it < (BR * KC) / (256 * 4); ++it) {
      int f = (tid + it * 256) << 2;             // float index in tile
      int r = f >> 6, c = f & 63;                // KC = 64
      uint32_t loff = sAoff + (uint32_t)((buf * (BR * LDA) + r * LDA + c) << 2);
      async_copy_b128(loff, L + (size_t)(rowBlk + r) * KDIM + kOff + c);
    }
    // M tiles (KC x CC): load float4, store transposed [n][k]
#pragma unroll
    for (int m = 0; m < NMAT; ++m) {
      const float* Msrc = (m == 0) ? M0 : M1;
#pragma unroll
      for (int it = 0; it < (KC * CC) / (256 * 4); ++it) {
        int f = (tid + it * 256) << 2;
        int r = f >> 5, c = f & 31;              // CC = 32
        float4 v = *(const float4*)(Msrc + (size_t)(kOff + r) * CC + c);
        float* B = &sBT[buf][m][0];
        B[(c + 0) * LDK + r] = v.x;
        B[(c + 1) * LDK + r] = v.y;
        B[(c + 2) * LDK + r] = v.z;
        B[(c + 3) * LDK + r] = v.w;
      }
    }
  };

  stage(0, kBase);
  wait_async0();
  __syncthreads();

  for (int kc = 0; kc < KEND; kc += KC) {
    const int cur = (kc / KC) & 1;
    if (kc + KC < KEND) stage(cur ^ 1, kBase + kc + KC);   // overlap next chunk

    const int arow = (wave << 4) | mrow;
#pragma unroll
    for (int kk = 0; kk < KC; kk += 4) {
      v2f a = *(const v2f*)&sA[cur][arow * LDA + kk + khalf];
      const int bk = kk + khalf;
#pragma unroll
      for (int m = 0; m < NMAT; ++m) {
        v2f b0 = *(const v2f*)&sBT[cur][m][bn * LDK + bk];
        v2f b1 = *(const v2f*)&sBT[cur][m][(bn + 16) * LDK + bk];
        acc[m][0] = wmma_f32(a, b0, acc[m][0]);
        acc[m][1] = wmma_f32(a, b1, acc[m][1]);
      }
    }
    wait_async0();          // next buffer's async copies done
    __syncthreads();        // all waves' LDS writes visible
  }

  // write partial sums (C/D layout: VGPR r -> rows r and r+8)
  float* outs[2] = {P0, P1};
  const int col  = lane & 15;
  const int rAdd = (lane >> 4) << 3;
#pragma unroll
  for (int m = 0; m < NMAT; ++m) {
    float* O = outs[m] + (size_t)blockIdx.y * SZ;
#pragma unroll
    for (int r = 0; r < 8; ++r) {
      size_t row = (size_t)rowBlk + (wave << 4) + rAdd + r;
      O[row * CC + col]      = acc[m][0][r];
      O[row * CC + 16 + col] = acc[m][1][r];
    }
  }
}

// Y = alpha * sum_k part[k] + beta * P   (deterministic K-split reduction)
__global__ void __launch_bounds__(256)
reduce_k(const float* __restrict__ part, const float* __restrict__ P,
         float* __restrict__ Y, float alpha, float beta) {
  int i = blockIdx.x * 256 + threadIdx.x;
  float s = 0.f;
#pragma unroll
  for (int k = 0; k < KSPLIT; ++k) s += part[(size_t)k * SZ + i];
  float y = alpha * s;
  if (P) y += beta * P[i];
  Y[i] = y;
}

// ------------------------------------------------------------------
// gates_k: Z = sigmoid(sum_k TX_k@Wxz[k] + TH_k@Whz[k] + b), R likewise,
// G = H * R.   One wave computes a 16x32 tile of Z and of R via WMMA.
// ------------------------------------------------------------------
__global__ void __launch_bounds__(64)
gates_k(const float* __restrict__ X,  const float* __restrict__ TX1, const float* __restrict__ TX2,
        const float* __restrict__ H,  const float* __restrict__ TH1, const float* __restrict__ TH2,
        const float* __restrict__ Wxz, const float* __restrict__ bxz,
        const float* __restrict__ Whz, const float* __restrict__ bhz,
        const float* __restrict__ Wxr, const float* __restrict__ bxr,
        const float* __restrict__ Whr, const float* __restrict__ bhr,
        float* __restrict__ Z, float* __restrict__ G) {
  const int lane  = threadIdx.x & 31;
  const int wave  = threadIdx.x >> 5;
  const int r0    = blockIdx.x * 32 + wave * 16;
  const int mrow  = lane & 15;
  const int khalf = (lane >> 4) << 1;
  const int bn    = lane & 15;

  const float* S[6]  = {X, TX1, TX2, H, TH1, TH2};
  const float* WZ[6] = {Wxz, Wxz + 1024, Wxz + 2048, Whz, Whz + 1024, Whz + 2048};
  const float* WR[6] = {Wxr, Wxr + 1024, Wxr + 2048, Whr, Whr + 1024, Whr + 2048};

  v8f aZ0 = {}, aZ1 = {}, aR0 = {}, aR1 = {};
#pragma unroll
  for (int s = 0; s < 6; ++s) {
#pragma unroll
    for (int kk = 0; kk < 32; kk += 4) {
      v2f a = *(const v2f*)&S[s][(size_t)(r0 + mrow) * CC + kk + khalf];
      const int bk = kk + khalf;
      v2f bz0, bz1, br0, br1;
      bz0.x = WZ[s][bk * 32 + bn];      bz0.y = WZ[s][(bk + 1) * 32 + bn];
      bz1.x = WZ[s][bk * 32 + 16 + bn]; bz1.y = WZ[s][(bk + 1) * 32 + 16 + bn];
      br0.x = WR[s][bk * 32 + bn];      br0.y = WR[s][(bk + 1) * 32 + bn];
      br1.x = WR[s][bk * 32 + 16 + bn]; br1.y = WR[s][(bk + 1) * 32 + 16 + bn];
      aZ0 = wmma_f32(a, bz0, aZ0);
      aZ1 = wmma_f32(a, bz1, aZ1);
      aR0 = wmma_f32(a, br0, aR0);
      aR1 = wmma_f32(a, br1, aR1);
    }
  }
  const int col  = lane & 15;
  const int rAdd = (lane >> 4) << 3;
  const float bZa = bxz[col] + bhz[col],      bZb = bxz[16 + col] + bhz[16 + col];
  const float bRa = bxr[col] + bhr[col],      bRb = bxr[16 + col] + bhr[16 + col];
#pragma unroll
  for (int r = 0; r < 8; ++r) {
    size_t row = (size_t)r0 + rAdd + r;
    float z0  = 1.f / (1.f + __expf(-(aZ0[r] + bZa)));
    float z1  = 1.f / (1.f + __expf(-(aZ1[r] + bZb)));
    float rr0 = 1.f / (1.f + __expf(-(aR0[r] + bRa)));
    float rr1 = 1.f / (1.f + __expf(-(aR1[r] + bRb)));
    Z[row * CC + col]      = z0;
    Z[row * CC + 16 + col] = z1;
    G[row * CC + col]      = H[row * CC + col] * rr0;
    G[row * CC + 16 + col] = H[row * CC + 16 + col] * rr1;
  }
}

// ------------------------------------------------------------------
// final_k: Htilde = tanh(sum_k TX_k@Wxh[k] + TG_k@Whh[k] + b)
//          out = Z*H + (1-Z)*Htilde
// ------------------------------------------------------------------
__global__ void __launch_bounds__(64)
final_k(const float* __restrict__ X,  const float* __restrict__ TX1, const float* __restrict__ TX2,
        const float* __restrict__ G,  const float* __restrict__ TG1, const float* __restrict__ TG2,
        const float* __restrict__ Wxh, const float* __restrict__ bxh,
        const float* __restrict__ Whh, const float* __restrict__ bhh,
        const float* __restrict__ Z,  const float* __restrict__ H,
        float* __restrict__ out) {
  const int lane  = threadIdx.x & 31;
  const int wave  = threadIdx.x >> 5;
  const int r0    = blockIdx.x * 32 + wave * 16;
  const int mrow  = lane & 15;
  const int khalf = (lane >> 4) << 1;
  const int bn    = lane & 15;

  const float* S[6] = {X, TX1, TX2, G, TG1, TG2};
  const float* W[6] = {Wxh, Wxh + 1024, Wxh + 2048, Whh, Whh + 1024, Whh + 2048};

  v8f a0 = {}, a1 = {};
#pragma unroll
  for (int s = 0; s < 6; ++s) {
#pragma unroll
    for (int kk = 0; kk < 32; kk += 4) {
      v2f a = *(const v2f*)&S[s][(size_t)(r0 + mrow) * CC + kk + khalf];
      const int bk = kk + khalf;
      v2f b0, b1;
      b0.x = W[s][bk * 32 + bn];      b0.y = W[s][(bk + 1) * 32 + bn];
      b1.x = W[s][bk * 32 + 16 + bn]; b1.y = W[s][(bk + 1) * 32 + 16 + bn];
      a0 = wmma_f32(a, b0, a0);
      a1 = wmma_f32(a, b1, a1);
    }
  }
  const int col  = lane & 15;
  const int rAdd = (lane >> 4) << 3;
  const float bHa = bxh[col] + bhh[col], bHb = bxh[16 + col] + bhh[16 + col];
#pragma unroll
  for (int r = 0; r < 8; ++r) {
    size_t row = (size_t)r0 + rAdd + r;
    float ht0 = tanhf(a0[r] + bHa);
    float ht1 = tanhf(a1[r] + bHb);
    float z0 = Z[row * CC + col],      z1 = Z[row * CC + 16 + col];
    float h0 = H[row * CC + col],      h1 = H[row * CC + 16 + col];
    out[row * CC + col]      = z0 * h0 + (1.f - z0) * ht0;
    out[row * CC + 16 + col] = z1 * h1 + (1.f - z1) * ht1;
  }
}

// ------------------------------------------------------------------
extern "C" void kernel_launch(void* const* d_in, const int* in_sizes, int n_in,
                              void* d_out, int out_size, void* d_ws, size_t ws_size,
                              hipStream_t stream) {
  const float* X   = (const float*)d_in[0];
  const float* L   = (const float*)d_in[1];
  const float* H   = (const float*)d_in[2];
  const float* Wxz = (const float*)d_in[3];  const float* bxz = (const float*)d_in[4];
  const float* Whz = (const float*)d_in[5];  const float* bhz = (const float*)d_in[6];
  const float* Wxr = (const float*)d_in[7];  const float* bxr = (const float*)d_in[8];
  const float* Whr = (const float*)d_in[9];  const float* bhr = (const float*)d_in[10];
  const float* Wxh = (const float*)d_in[11]; const float* bxh = (const float*)d_in[12];
  const float* Whh = (const float*)d_in[13]; const float* bhh = (const float*)d_in[14];

  float* ws    = (float*)d_ws;
  float* TX1   = ws + 0 * (size_t)SZ;
  float* TX2   = ws + 1 * (size_t)SZ;
  float* TH1   = ws + 2 * (size_t)SZ;
  float* TH2   = ws + 3 * (size_t)SZ;
  float* G     = ws + 4 * (size_t)SZ;
  float* TG1   = ws + 5 * (size_t)SZ;
  float* TG2   = ws + 6 * (size_t)SZ;
  float* Zb    = ws + 7 * (size_t)SZ;
  float* part0 = ws + 8 * (size_t)SZ;
  float* part1 = ws + (size_t)(8 + KSPLIT) * SZ;
  float* out   = (float*)d_out;

  dim3 gB(NROW / BR, KSPLIT);   // 64 x 4 blocks, 8 waves each
  dim3 gR(SZ / 256);
  dim3 gE(NROW / 32);

  // pass 1: TX1 = L@X, TH1 = L@H  (shared L traffic)
  bigmm<2><<<gB, 256, 0, stream>>>(L, X, H, part0, part1);
  reduce_k<<<gR, 256, 0, stream>>>(part0, nullptr, TX1, 1.f, 0.f);
  reduce_k<<<gR, 256, 0, stream>>>(part1, nullptr, TH1, 1.f, 0.f);

  // pass 2: TX2 = 2 L@TX1 - X, TH2 = 2 L@TH1 - H
  bigmm<2><<<gB, 256, 0, stream>>>(L, TX1, TH1, part0, part1);
  reduce_k<<<gR, 256, 0, stream>>>(part0, X, TX2, 2.f, -1.f);
  reduce_k<<<gR, 256, 0, stream>>>(part1, H, TH2, 2.f, -1.f);

  // gates: Z, G = H * R
  gates_k<<<gE, 64, 0, stream>>>(X, TX1, TX2, H, TH1, TH2,
                                 Wxz, bxz, Whz, bhz, Wxr, bxr, Whr, bhr, Zb, G);

  // pass 3: TG1 = L@G
  bigmm<1><<<gB, 256, 0, stream>>>(L, G, G, part0, part0);
  reduce_k<<<gR, 256, 0, stream>>>(part0, nullptr, TG1, 1.f, 0.f);

  // pass 4: TG2 = 2 L@TG1 - G
  bigmm<1><<<gB, 256, 0, stream>>>(L, TG1, TG1, part0, part0);
  reduce_k<<<gR, 256, 0, stream>>>(part0, G, TG2, 2.f, -1.f);

  // epilogue: Htilde + GRU blend
  final_k<<<gE, 64, 0, stream>>>(X, TX1, TX2, G, TG1, TG2,
                                 Wxh, bxh, Whh, bhh, Zb, H, out);
}